// DeformableConv2d_66589172957393
// MI455X (gfx1250) — compile-verified
//
#include <hip/hip_runtime.h>

typedef __attribute__((ext_vector_type(2))) float v2f;
typedef __attribute__((ext_vector_type(8))) float v8f;
typedef __attribute__((ext_vector_type(4))) unsigned int u32x4;
typedef __attribute__((ext_vector_type(8))) int i32x8;
typedef __attribute__((ext_vector_type(4))) int i32x4;

#define Bn   8
#define Cc   64
#define Oo   64
#define Hh   128
#define Ww   128
#define HW   (Hh * Ww)
#define TAPS 9
#define KTOT (Cc * TAPS)          // 576
#define CH_BYTES (HW * 4)         // 65536 bytes per channel plane
#define WROW 80                   // LDS weight-tile row stride (dwords): 64 data + 16 pad

#if defined(__HIP_DEVICE_COMPILE__) && __has_builtin(__builtin_amdgcn_tensor_load_to_lds)
#define USE_TDM 1
#else
#define USE_TDM 0
#endif

// ---------------------------------------------------------------------------
// Kernel 0: transpose w_conv[O][C][3][3] -> wT[k][O] with k = tap*C + c.
// ---------------------------------------------------------------------------
__global__ __launch_bounds__(256) void k_wt_transpose(const float* __restrict__ w_conv,
                                                      float* __restrict__ wT) {
    int i = blockIdx.x * 256 + threadIdx.x;      // over KTOT*Oo = 36864
    if (i >= KTOT * Oo) return;
    int o = i & 63;
    int k = i >> 6;
    int t = k / Cc;
    int c = k % Cc;
    wT[k * Oo + o] = w_conv[o * KTOT + c * TAPS + t];
}

// ---------------------------------------------------------------------------
// Kernel 1: offset conv (N=18, fp32 direct). j uniform per block.
// ---------------------------------------------------------------------------
__global__ __launch_bounds__(256) void k_offset_conv(const float* __restrict__ x,
                                                     const float* __restrict__ w_off,
                                                     const float* __restrict__ b_off,
                                                     float* __restrict__ off) {
    int chunk = blockIdx.x & 63;                 // 16384/256 = 64 chunks
    int bj    = blockIdx.x >> 6;
    int j     = bj % 18;
    int b     = bj / 18;
    int pix   = chunk * 256 + threadIdx.x;
    int h = pix >> 7, w = pix & 127;

    const float* xb = x + (size_t)b * Cc * HW;
    const float* wj = w_off + j * KTOT;          // [c][ky][kx]
    float acc = b_off[j];

    for (int c = 0; c < Cc; ++c) {
        const float* xc = xb + c * HW;
        const float* wc = wj + c * 9;
#pragma unroll
        for (int ky = 0; ky < 3; ++ky) {
            int y = h + ky - 1;
            if ((unsigned)y < (unsigned)Hh) {
#pragma unroll
                for (int kx = 0; kx < 3; ++kx) {
                    int xw = w + kx - 1;
                    if ((unsigned)xw < (unsigned)Ww)
                        acc = fmaf(xc[y * Ww + xw], wc[ky * 3 + kx], acc);
                }
            }
        }
    }
    off[((size_t)b * 18 + j) * HW + pix] = acc;
}

// ---------------------------------------------------------------------------
// Kernel 2: bilinear gather + implicit GEMM on V_WMMA_F32_16X16X4_F32.
// Per tap: weight tile (64K x 64O = 16KB) staged into LDS via the Tensor
// Data Mover (pad feature gives an 80-dword row stride -> bank-conflict-free
// B-fragment reads), synchronized with s_wait_tensorcnt.
// ---------------------------------------------------------------------------
__global__ __launch_bounds__(128) void k_deform_wmma(const float* __restrict__ x,
                                                     const float* __restrict__ off,
                                                     const float* __restrict__ wT,
                                                     float* __restrict__ out) {
    __shared__ __align__(16) int   s_off[64 * TAPS * 4];   // corner byte offsets
    __shared__ __align__(16) float s_w  [64 * TAPS * 4];   // corner weights (0 if OOB)
    __shared__ __align__(16) float s_wt [Cc * WROW];       // per-tap weight tile (padded rows)

    const int tid  = threadIdx.x;
    const int wave = tid >> 5;
    const int lane = tid & 31;
    const int b        = blockIdx.x >> 8;         // 256 tiles per batch
    const int tileBase = (blockIdx.x & 255) * 64;

    const char* xbp = (const char*)(x + (size_t)b * Cc * HW);   // uniform batch base

    // ---- Phase A: per (pixel, tap) bilinear setup into LDS ----
    for (int i = lane; i < 16 * TAPS; i += 32) {
        int p = i / TAPS, t = i % TAPS;
        int pw  = wave * 16 + p;
        int pix = tileBase + pw;
        int h = pix >> 7, w = pix & 127;

        float dy = off[((size_t)b * 18 + t * 2 + 0) * HW + pix];
        float dx = off[((size_t)b * 18 + t * 2 + 1) * HW + pix];
        float py = (float)(h - 1 + t / 3) + dy;
        float px = (float)(w - 1 + t % 3) + dx;
        float y0f = floorf(py), x0f = floorf(px);
        float ly = py - y0f, lx = px - x0f;
        int iy0 = (int)y0f, ix0 = (int)x0f;

        int e4 = (pw * TAPS + t) * 4;
#pragma unroll
        for (int corner = 0; corner < 4; ++corner) {
            int ry = iy0 + (corner >> 1);
            int rx = ix0 + (corner & 1);
            bool valid = ((unsigned)ry < (unsigned)Hh) && ((unsigned)rx < (unsigned)Ww);
            int cy = min(max(ry, 0), Hh - 1);
            int cx = min(max(rx, 0), Ww - 1);
            s_off[e4 + corner] = (cy * Ww + cx) * 4;
            float wy = (corner >> 1) ? ly : 1.0f - ly;
            float wx = (corner & 1)  ? lx : 1.0f - lx;
            s_w[e4 + corner] = valid ? wy * wx : 0.0f;
        }
    }
    __syncthreads();

    // ---- Phase B/C: K-loop (tap-major, c within tap) with fp32 WMMA ----
    const int mrow  = lane & 15;       // A: row M; B/D: col N
    const int khalf = lane >> 4;       // 0 -> K{0,1}, 1 -> K{2,3}
    v8f acc[4] = {};                   // 4 N-tiles of 16 channels

    for (int t = 0; t < TAPS; ++t) {
        __syncthreads();               // previous tap's s_wt reads complete

        // ---- stage weight tile for this tap into LDS ----
#if USE_TDM
        if (wave == 0) {
            unsigned long long gaddr =
                (unsigned long long)(uintptr_t)wT + (unsigned long long)t * (Cc * Oo * 4);
            unsigned lds_off = (unsigned)(uintptr_t)(void*)s_wt;  // LDS byte offset
            u32x4 g0;
            g0.x = 1u;                                            // count=1 (valid user D#)
            g0.y = lds_off;                                       // lds_addr
            g0.z = (unsigned)gaddr;                               // global_addr[31:0]
            g0.w = (unsigned)((gaddr >> 32) & 0x01FFFFFFull) | (2u << 30);  // addr[56:32]|type=2
            i32x8 g1;
            g1[0] = (2 << 16) | (1 << 20) | (5 << 22) | (15 << 25); // 4B elems, pad 16dw/64dw
            g1[1] = (int)(4096u << 16);                           // tensor_dim0 = 4096
            g1[2] = (int)(1u << 16);                              // tensor_dim1 = 1
            g1[3] = (int)(4096u << 16);                           // tile_dim0 = 4096
            g1[4] = 1;                                            // tile_dim1 = 1
            g1[5] = 4096;                                         // tensor_dim0_stride lo
            g1[6] = (int)(4096u << 16);                           // tensor_dim1_stride lo
            g1[7] = 0;
            i32x4 g2 = {0, 0, 0, 0};
            i32x4 g3 = {0, 0, 0, 0};
#if __clang_major__ >= 23
            i32x8 g4 = {0, 0, 0, 0, 0, 0, 0, 0};
            __builtin_amdgcn_tensor_load_to_lds(g0, g1, g2, g3, g4, 0);
#else
            __builtin_amdgcn_tensor_load_to_lds(g0, g1, g2, g3, 0);
#endif
            __builtin_amdgcn_s_wait_tensorcnt(0);
        }
#else
        {
            const float* wTt = wT + t * (Cc * Oo);
            for (int i = tid; i < Cc * Oo; i += 128)
                s_wt[(i >> 6) * WROW + (i & 63)] = wTt[i];
        }
#endif
        __syncthreads();

        int e4 = ((wave * 16 + mrow) * TAPS + t) * 4;
        int4   co = *(const int4*)  &s_off[e4];
        float4 cw = *(const float4*)&s_w[e4];

        for (int c0 = 0; c0 < Cc; c0 += 4) {
            int c = c0 + khalf * 2;
            unsigned cb = (unsigned)c * CH_BYTES;   // u32 per-lane offset -> SADDR+voffset form
            unsigned o0 = cb + (unsigned)co.x;
            unsigned o1 = cb + (unsigned)co.y;
            unsigned o2 = cb + (unsigned)co.z;
            unsigned o3 = cb + (unsigned)co.w;

            // A fragment: lane holds (M=mrow, K=2*khalf+{0,1}) -> channels c, c+1
            float a0 = cw.x * *(const float*)(xbp + (size_t)o0)
                     + cw.y * *(const float*)(xbp + (size_t)o1)
                     + cw.z * *(const float*)(xbp + (size_t)o2)
                     + cw.w * *(const float*)(xbp + (size_t)o3);
            float a1 = cw.x * *(const float*)(xbp + (size_t)o0 + CH_BYTES)
                     + cw.y * *(const float*)(xbp + (size_t)o1 + CH_BYTES)
                     + cw.z * *(const float*)(xbp + (size_t)o2 + CH_BYTES)
                     + cw.w * *(const float*)(xbp + (size_t)o3 + CH_BYTES);
            v2f A; A.x = a0; A.y = a1;

            const float* wr = s_wt + c * WROW;      // LDS rows K=c, c+1 (bank-conflict-free)
#pragma unroll
            for (int n = 0; n < 4; ++n) {
                int o = n * 16 + mrow;
                v2f Bf;
                Bf.x = wr[o];                       // K = 2*khalf
                Bf.y = wr[WROW + o];                // K = 2*khalf + 1
                acc[n] = __builtin_amdgcn_wmma_f32_16x16x4_f32(
                    /*neg_a=*/false, A, /*neg_b=*/false, Bf,
                    /*c_mod=*/(short)0, acc[n],
                    /*reuse_a=*/false, /*reuse_b=*/false);
            }
        }
    }

    // ---- Phase D: store. D VGPR g: M = g + 8*khalf, N = mrow ----
#pragma unroll
    for (int n = 0; n < 4; ++n) {
        int o = n * 16 + mrow;
        float* op = out + ((size_t)b * Oo + o) * HW + tileBase + wave * 16 + 8 * khalf;
#pragma unroll
        for (int g = 0; g < 8; ++g) op[g] = acc[n][g];
    }
}

// ---------------------------------------------------------------------------
extern "C" void kernel_launch(void* const* d_in, const int* in_sizes, int n_in,
                              void* d_out, int out_size, void* d_ws, size_t ws_size,
                              hipStream_t stream) {
    (void)in_sizes; (void)n_in; (void)out_size; (void)ws_size;
    const float* x      = (const float*)d_in[0];
    const float* w_off  = (const float*)d_in[1];
    const float* b_off  = (const float*)d_in[2];
    const float* w_conv = (const float*)d_in[3];
    float* out = (float*)d_out;

    float* wT     = (float*)d_ws;                                   // 147456 B
    float* offbuf = (float*)((char*)d_ws + (size_t)KTOT * Oo * 4);  // 9.4 MB

    k_wt_transpose<<<(KTOT * Oo + 255) / 256, 256, 0, stream>>>(w_conv, wT);
    k_offset_conv<<<Bn * 18 * 64, 256, 0, stream>>>(x, w_off, b_off, offbuf);
    k_deform_wmma<<<Bn * 256, 128, 0, stream>>>(x, offbuf, wT, out);
}